// Propagation_67963562492185
// MI455X (gfx1250) — compile-verified
//
#include <hip/hip_runtime.h>

typedef __attribute__((ext_vector_type(2))) float v2f;
typedef __attribute__((ext_vector_type(8))) float v8f;

#define D_FEAT 64

// ---------------------------------------------------------------------------
// Zero the output buffer (harness poisons d_out with 0xAA).
// ---------------------------------------------------------------------------
__global__ void __launch_bounds__(256)
zero_kernel(float* __restrict__ out, int n) {
    int i4 = (blockIdx.x * blockDim.x + threadIdx.x) * 4;
    if (i4 + 3 < n) {
        *(float4*)(out + i4) = make_float4(0.f, 0.f, 0.f, 0.f);
    } else {
        for (int k = i4; k < n; ++k) out[k] = 0.f;
    }
}

// ---------------------------------------------------------------------------
// WMMA-based COO SpMM scatter.
// One wave32 handles one group of 16 edges.
//   msgs[M][:] = v[base+M] * H[cols[base+M]][:]   (M = 0..15, 64 features)
// computed as  D = diag(v) x B  via 4 chained V_WMMA_F32_16X16X4_F32 per
// 16-feature tile (full f32 precision), then scattered with
// global_atomic_add_f32 into out[rows[base+M]][:].
// ---------------------------------------------------------------------------
__global__ void __launch_bounds__(256)
spmm_wmma_kernel(const float* __restrict__ H,
                 const float* __restrict__ vals,
                 const int*   __restrict__ rows,
                 const int*   __restrict__ cols,
                 float*       __restrict__ out,
                 int ngroups) {
    const int lane = threadIdx.x & 31;
    const int wave = (int)((blockIdx.x * blockDim.x + threadIdx.x) >> 5);
    if (wave >= ngroups) return;          // whole wave exits together

    const long base  = (long)wave * 16;   // first edge of this group
    const int  m     = lane & 15;         // edge-in-group / column-in-tile
    const int  khalf = (lane >> 4) << 1;  // 0 for lanes 0-15, 2 for lanes 16-31

    // Per-edge scalars, distributed one edge per lane (duplicated in upper half).
    const float vm = vals[base + m];
    const int   rm = rows[base + m];
    const int   cm = cols[base + m];

    // A chunks: 16x4 slices of diag(v).
    // ISA layout (32-bit A 16x4): lane L -> M = L&15; VGPR0 holds local K =
    // (L<16 ? 0 : 2), VGPR1 holds local K+1. Chunk q covers global K = 4q..4q+3.
    v2f a[4];
#pragma unroll
    for (int q = 0; q < 4; ++q) {
        const int k0 = 4 * q + khalf;
        const int k1 = k0 + 1;
        a[q].x = (m == k0) ? vm : 0.0f;
        a[q].y = (m == k1) ? vm : 0.0f;
    }

#pragma unroll
    for (int t = 0; t < 4; ++t) {         // four 16-wide feature tiles
        v8f acc = {0.f, 0.f, 0.f, 0.f, 0.f, 0.f, 0.f, 0.f};
#pragma unroll
        for (int q = 0; q < 4; ++q) {     // K-chunks of the 16x16 diag product
            const int K0 = 4 * q + khalf;
            const int K1 = K0 + 1;
            // B layout (4x16): lane L -> N = L&15; VGPR0 = row K0, VGPR1 = row K1.
            const int c0 = __shfl(cm, K0, 32);
            const int c1 = __shfl(cm, K1, 32);
            v2f b;
            b.x = H[(long)c0 * D_FEAT + t * 16 + m];  // lanes 0-15: 64B coalesced
            b.y = H[(long)c1 * D_FEAT + t * 16 + m];
            acc = __builtin_amdgcn_wmma_f32_16x16x4_f32(
                /*neg_a=*/false, a[q], /*neg_b=*/false, b,
                /*c_mod=*/(short)0, acc, /*reuse_a=*/false, /*reuse_b=*/false);
        }
        // D layout (32-bit C/D 16x16): VGPR j -> M = j + (lane>=16 ? 8 : 0),
        // N = lane&15. Scatter each element with an f32 atomic (lands in L2).
#pragma unroll
        for (int j = 0; j < 8; ++j) {
            const int M = j + ((lane >> 4) << 3);
            const int r = __shfl(rm, M, 32);
            atomicAdd(&out[(long)r * D_FEAT + t * 16 + m], acc[j]);
        }
    }
}

// ---------------------------------------------------------------------------
// Scalar tail for E % 16 edges (E = 1.6M is a multiple of 16, so normally
// this never launches; kept for generality).
// ---------------------------------------------------------------------------
__global__ void __launch_bounds__(256)
spmm_tail_kernel(const float* __restrict__ H,
                 const float* __restrict__ vals,
                 const int*   __restrict__ rows,
                 const int*   __restrict__ cols,
                 float*       __restrict__ out,
                 long start, long nwork) {
    long t = (long)blockIdx.x * blockDim.x + threadIdx.x;
    if (t >= nwork) return;
    long e = start + t / D_FEAT;
    int  f = (int)(t % D_FEAT);
    atomicAdd(&out[(long)rows[e] * D_FEAT + f],
              vals[e] * H[(long)cols[e] * D_FEAT + f]);
}

extern "C" void kernel_launch(void* const* d_in, const int* in_sizes, int n_in,
                              void* d_out, int out_size, void* d_ws, size_t ws_size,
                              hipStream_t stream) {
    const float* H    = (const float*)d_in[0];
    const float* vals = (const float*)d_in[1];
    const int*   rows = (const int*)d_in[2];
    const int*   cols = (const int*)d_in[3];
    float*       out  = (float*)d_out;

    const long E = in_sizes[1];

    // 1) zero the (poisoned) output
    {
        int nthreads = (out_size + 3) / 4;
        int grid = (nthreads + 255) / 256;
        zero_kernel<<<grid, 256, 0, stream>>>(out, out_size);
    }

    // 2) WMMA scatter over full 16-edge groups (one group per wave32)
    const long ngroups = E / 16;
    if (ngroups > 0) {
        long threads = ngroups * 32;
        long grid = (threads + 255) / 256;
        spmm_wmma_kernel<<<(unsigned)grid, 256, 0, stream>>>(
            H, vals, rows, cols, out, (int)ngroups);
    }

    // 3) scalar tail (E % 16 edges)
    const long tail = E - ngroups * 16;
    if (tail > 0) {
        long nwork = tail * D_FEAT;
        long grid = (nwork + 255) / 256;
        spmm_tail_kernel<<<(unsigned)grid, 256, 0, stream>>>(
            H, vals, rows, cols, out, ngroups * 16, nwork);
    }
}